// WongWangMultiClassDecision_88227218195048
// MI455X (gfx1250) — compile-verified
//
#include <hip/hip_runtime.h>

// Wong-Wang multi-class decision, CDNA5 (gfx1250, wave32).
// Layout: one wave = 16 batch rows. lane k (0-15) owns channels 0-3 of row k,
// lane 16+k owns channels 4-7 of row k. The 8x8 coupling matmul is computed as
// (s @ J)^T = J^T * s^T via two V_WMMA_F32_16X16X4_F32 (K=8 split into 2x K=4).
// All cross-half data motion is a lane^16 exchange (__shfl_xor).

#define TSTEPS 1000
#define BSZ    4096
#define CH     8

typedef float v2f __attribute__((ext_vector_type(2)));
typedef float v8f __attribute__((ext_vector_type(8)));

__global__ __launch_bounds__(256) void wongwang_wmma_kernel(
    const float* __restrict__ x,     // [B, C]
    const float* __restrict__ eps0,  // [B, C]
    const float* __restrict__ eps,   // [T, B, C]
    const float* __restrict__ Jm,    // [C, C] row-major: Jm[kin*8 + mout]
    const float* __restrict__ jext_p,
    const float* __restrict__ i0_p,
    const float* __restrict__ na_p,
    const float* __restrict__ thr_p,
    float* __restrict__ out)         // [B, C]
{
  const int tid  = blockIdx.x * blockDim.x + threadIdx.x;
  const int lane = threadIdx.x & 31;
  const int k    = lane & 15;   // batch row within the wave tile
  const int h    = lane >> 4;   // 0: channels 0-3, 1: channels 4-7
  const int wave = tid >> 5;
  const int row  = wave * 16 + k;

  // Uniform scalar parameters (single-element device arrays).
  const float J_ext = jext_p[0];
  const float I0    = i0_p[0];
  const float na    = na_p[0];
  const float thr   = thr_p[0];

  const float Acoef = 270.0f;
  const float Bcoef = 108.0f;
  const float Dcoef = 0.154f;
  const float decay  = __expf(-0.25f);                                  // exp(-dt/tau_ampa)
  const float nscale = na * __builtin_sqrtf((1.0f - __expf(-0.5f)) * 0.5f);

  // ---- A tiles: J^T in WMMA 32-bit A layout (16x4, M=out channel, K=in channel).
  // A VGPR0 holds K = 2h+0, VGPR1 holds K = 2h+1 (tile 1: K=0..3); tile 2: K=4..7.
  // Rows M=8..15 of A are zero-padded -> D for lanes 16-31 is exactly zero.
  const int   mm    = k & 7;
  const float valid = (k < 8) ? 1.0f : 0.0f;
  v2f a1, a2;
  a1.x = Jm[(2*h + 0)*CH + mm] * valid;   // A[m][2h+0] = J[2h+0][m]
  a1.y = Jm[(2*h + 1)*CH + mm] * valid;
  a2.x = Jm[(2*h + 4)*CH + mm] * valid;
  a2.y = Jm[(2*h + 5)*CH + mm] * valid;

  // ---- Per-lane state: 4 channels of one batch row, all in registers.
  const int base = row * CH + h * 4;
  const float4 xv = *(const float4*)(x + base);
  const float4 e0 = *(const float4*)(eps0 + base);

  float s[4]   = {0.1f, 0.1f, 0.1f, 0.1f};
  float In[4]  = {na * e0.x, na * e0.y, na * e0.z, na * e0.w};
  float Iex[4] = {J_ext * xv.x, J_ext * xv.y, J_ext * xv.z, J_ext * xv.w};
  int   dec[4] = {-1, -1, -1, -1};

  // ---- Time loop: stream eps with a 1-step register pipeline + deep L2 prefetch
  // (whole eps slice nearly fits the 192MB L2; 23.3 TB/s floor is ~5.6us).
  const float* pe = eps + base;
  float4 ev = *(const float4*)pe;

  for (int t = 0; t < TSTEPS; ++t) {
    const int tn = (t + 1 < TSTEPS) ? (t + 1) : t;
    const float4 evn = *(const float4*)(pe + (size_t)tn * (BSZ * CH));
    const int tp = (t + 16 < TSTEPS) ? (t + 16) : (TSTEPS - 1);
    __builtin_prefetch(pe + (size_t)tp * (BSZ * CH), 0, 3);   // global_prefetch_b8

    // ---- Build B tiles (s^T, 4x16). Every exchange is lane^16.
    // B VGPR0: lanes 0-15 -> (K=0,N=lane), lanes 16-31 -> (K=2,N=lane-16); VGPR1: K=1/K=3.
    const float s2k = __shfl_xor(s[2], 16, 32);   // upper half fetches lower's ch2
    const float s3k = __shfl_xor(s[3], 16, 32);   // upper half fetches lower's ch3
    const float s0h = __shfl_xor(s[0], 16, 32);   // lower half fetches upper's ch4
    const float s1h = __shfl_xor(s[1], 16, 32);   // lower half fetches upper's ch5
    v2f b1, b2;
    b1.x = (h == 0) ? s[0] : s2k;   // K=0 / K=2
    b1.y = (h == 0) ? s[1] : s3k;   // K=1 / K=3
    b2.x = (h == 0) ? s0h  : s[2];  // K=4 / K=6
    b2.y = (h == 0) ? s1h  : s[3];  // K=5 / K=7

    // ---- D = J^T * s^T : lane N (0-15) gets channel r of row N in acc[r].
    v8f acc = {0.f, 0.f, 0.f, 0.f, 0.f, 0.f, 0.f, 0.f};
    acc = __builtin_amdgcn_wmma_f32_16x16x4_f32(false, a1, false, b1,
                                                (short)0, acc, false, false);
    acc = __builtin_amdgcn_wmma_f32_16x16x4_f32(false, a2, false, b2,
                                                (short)0, acc, false, false);

    // Redistribute: lanes 16-31 need channels 4-7 (lanes 0-15, VGPRs 4-7) = lane^16.
    float d[4];
#pragma unroll
    for (int c = 0; c < 4; ++c) {
      const float hi = __shfl_xor(acc[c + 4], 16, 32);
      d[c] = (h == 0) ? acc[c] : hi;
    }

    const float e[4] = {ev.x, ev.y, ev.z, ev.w};
#pragma unroll
    for (int c = 0; c < 4; ++c) {
      const float u  = Acoef * (d[c] + I0 + Iex[c] + In[c]) - Bcoef;
      const float ex = __expf(-Dcoef * u);
      const float den = 1.0f - ex + 1e-6f;
      // v_rcp_f32 + mul instead of IEEE v_div_scale/div_fmas sequence (~10 VALU saved/ch).
      float Hn = u * __builtin_amdgcn_rcpf(den);
      Hn = fmaxf(Hn, 0.0f);
      const float dsdt = -s[c] * 0.01f + (1.0f - s[c]) * Hn * 0.000641f; // gamma/1000
      In[c] = In[c] * decay + nscale * e[c];
      s[c]  = fmaf(dsdt, 0.5f, s[c]);                                    // dt = 0.5
      dec[c] = (dec[c] < 0 && s[c] > thr) ? t : dec[c];                  // first crossing
    }
    ev = evn;
  }

  // out = dec * DT / 1000 (never crossed -> T-1)
  float4 o;
  o.x = (float)((dec[0] < 0) ? (TSTEPS - 1) : dec[0]) * 5.0e-4f;
  o.y = (float)((dec[1] < 0) ? (TSTEPS - 1) : dec[1]) * 5.0e-4f;
  o.z = (float)((dec[2] < 0) ? (TSTEPS - 1) : dec[2]) * 5.0e-4f;
  o.w = (float)((dec[3] < 0) ? (TSTEPS - 1) : dec[3]) * 5.0e-4f;
  *(float4*)(out + base) = o;
}

extern "C" void kernel_launch(void* const* d_in, const int* in_sizes, int n_in,
                              void* d_out, int out_size, void* d_ws, size_t ws_size,
                              hipStream_t stream) {
  (void)in_sizes; (void)n_in; (void)out_size; (void)d_ws; (void)ws_size;
  const float* x    = (const float*)d_in[0];
  const float* eps0 = (const float*)d_in[1];
  const float* eps  = (const float*)d_in[2];
  const float* Jm   = (const float*)d_in[3];
  const float* jext = (const float*)d_in[4];
  const float* i0   = (const float*)d_in[5];
  const float* na   = (const float*)d_in[6];
  const float* thr  = (const float*)d_in[7];
  float* out = (float*)d_out;

  // 4096 rows / 16 rows-per-wave = 256 waves = 8192 threads.
  dim3 grid(32), block(256);
  hipLaunchKernelGGL(wongwang_wmma_kernel, grid, block, 0, stream,
                     x, eps0, eps, Jm, jext, i0, na, thr, out);
}